// GAT_80977313399736
// MI455X (gfx1250) — compile-verified
//
#include <hip/hip_runtime.h>
#include <hip/hip_bf16.h>
#include <math.h>

typedef float v2f __attribute__((ext_vector_type(2)));
typedef float v8f __attribute__((ext_vector_type(8)));
typedef int   v2i __attribute__((vector_size(2 * sizeof(int))));
typedef int   v4i __attribute__((vector_size(4 * sizeof(int))));

#define NEG_SLOPE 0.2f
#define TILE_K 32

// gfx1250 async-to-LDS path (ASYNCcnt-tracked DMA into LDS), guarded so a
// missing builtin degrades to a synchronous double-buffered fallback.
#if defined(__has_builtin)
#if __has_builtin(__builtin_amdgcn_global_load_async_to_lds_b128) && \
    __has_builtin(__builtin_amdgcn_global_load_async_to_lds_b64)  && \
    __has_builtin(__builtin_amdgcn_s_wait_asynccnt)
#define USE_ASYNC_LDS 1
#endif
#endif

// Builtin signature (from hipcc diagnostics): width-typed vector pointers,
// global src in addrspace(1) (prints as __device__), LDS dst in addrspace(3).
#define GPTR2(p) ((__attribute__((address_space(1))) v2i*)(p))
#define GPTR4(p) ((__attribute__((address_space(1))) v4i*)(p))
#define LPTR2(p) ((__attribute__((address_space(3))) v2i*)(p))
#define LPTR4(p) ((__attribute__((address_space(3))) v4i*)(p))

// ---------------------------------------------------------------------------
// FP32 WMMA GEMM: C[M,Nn] = A[M,K] * B[K,Nn], row-major, all dims divisible
// (M%16==0, K%32==0, Nn%128==0). Block = 256 thr = 8 waves; wave -> 16x16 tile.
// V_WMMA_F32_16X16X4_F32 on double-buffered LDS tiles filled by
// GLOBAL_LOAD_ASYNC_TO_LDS (5 async instructions per wave per stage).
// ---------------------------------------------------------------------------
__global__ __launch_bounds__(256)
void gemm_wmma_f32(const float* __restrict__ A, const float* __restrict__ B,
                   float* __restrict__ C, int M, int K, int Nn) {
  __shared__ __align__(16) float As[2][16][TILE_K];    // 2 x 2 KB
  __shared__ __align__(16) float Bs[2][TILE_K][128];   // 2 x 16 KB
  const int tid  = threadIdx.x;
  const int wave = tid >> 5;          // 0..7
  const int lane = tid & 31;
  const int m0 = blockIdx.x * 16;
  const int n0 = blockIdx.y * 128;

  const int mrow = lane & 15;
  const int koff = (lane >> 4) * 2;   // 0 or 2
  const int ncol = wave * 16 + (lane & 15);

  // staging coordinates: A tile = 256 x B64 chunks (1/thread),
  //                      B tile = 1024 x B128 chunks (4/thread)
  const int ar = tid >> 4;            // 0..15
  const int ac = (tid & 15) * 2;      // 0,2,..,30

  auto stage = [&](int buf, int k0) {
#ifdef USE_ASYNC_LDS
    __builtin_amdgcn_global_load_async_to_lds_b64(
        GPTR2(&A[(size_t)(m0 + ar) * K + (k0 + ac)]),
        LPTR2(&As[buf][ar][ac]), 0, 0);
    #pragma unroll
    for (int i = 0; i < 4; ++i) {
      int idx = tid + i * 256;        // 0..1023
      int r = idx >> 5, c = (idx & 31) * 4;
      __builtin_amdgcn_global_load_async_to_lds_b128(
          GPTR4(&B[(size_t)(k0 + r) * Nn + (n0 + c)]),
          LPTR4(&Bs[buf][r][c]), 0, 0);
    }
#else
    *(float2*)&As[buf][ar][ac] =
        *(const float2*)&A[(size_t)(m0 + ar) * K + (k0 + ac)];
    #pragma unroll
    for (int i = 0; i < 4; ++i) {
      int idx = tid + i * 256;
      int r = idx >> 5, c = (idx & 31) * 4;
      *(float4*)&Bs[buf][r][c] =
          *(const float4*)&B[(size_t)(k0 + r) * Nn + (n0 + c)];
    }
#endif
  };

  v8f acc = {};
  stage(0, 0);
  int buf = 0;
  for (int k0 = 0; k0 < K; k0 += TILE_K) {
    const bool more = (k0 + TILE_K) < K;
    if (more) stage(buf ^ 1, k0 + TILE_K);   // prefetch next tile into other buffer
#ifdef USE_ASYNC_LDS
    // async loads retire in order: <=5 outstanding == previous stage complete
    if (more) __builtin_amdgcn_s_wait_asynccnt(5);
    else      __builtin_amdgcn_s_wait_asynccnt(0);
#endif
    __syncthreads();
    #pragma unroll
    for (int kk = 0; kk < TILE_K; kk += 4) {
      v2f a, b;
      a.x = As[buf][mrow][kk + koff];
      a.y = As[buf][mrow][kk + koff + 1];
      b.x = Bs[buf][kk + koff][ncol];
      b.y = Bs[buf][kk + koff + 1][ncol];
      acc = __builtin_amdgcn_wmma_f32_16x16x4_f32(
          /*neg_a=*/false, a, /*neg_b=*/false, b,
          /*c_mod=*/(short)0, acc, /*reuse_a=*/false, /*reuse_b=*/false);
    }
    __syncthreads();   // protects the buffer we are about to refill
    buf ^= 1;
  }

  // D layout: VGPR r -> row (m0 + r + (lane>=16 ? 8:0)), col = n0 + wave*16 + (lane&15)
  const int col   = n0 + ncol;
  const int rbase = m0 + ((lane >> 4) ? 8 : 0);
  #pragma unroll
  for (int r = 0; r < 8; ++r)
    C[(size_t)(rbase + r) * Nn + col] = acc[r];
}

// ---------------------------------------------------------------------------
// Attention coefficients: a_src[n,h] = <h[n,h,:], att_src[h,:]>, same for dst
// ---------------------------------------------------------------------------
__global__ void attn_coef(const float* __restrict__ h,
                          const float* __restrict__ att_src,
                          const float* __restrict__ att_dst,
                          float* __restrict__ a_src, float* __restrict__ a_dst,
                          int N, int H, int C) {
  int i = blockIdx.x * blockDim.x + threadIdx.x;
  if (i >= N * H) return;
  int n = i / H, hh = i - n * H;
  const float* hv = h + (size_t)n * H * C + (size_t)hh * C;
  const float* as = att_src + (size_t)hh * C;
  const float* ad = att_dst + (size_t)hh * C;
  float s1 = 0.f, s2 = 0.f;
  for (int c = 0; c < C; ++c) { float v = hv[c]; s1 += v * as[c]; s2 += v * ad[c]; }
  a_src[i] = s1;
  a_dst[i] = s2;
}

__global__ void fill_f32(float* __restrict__ p, float v, size_t n) {
  size_t i = (size_t)blockIdx.x * blockDim.x + threadIdx.x;
  if (i < n) p[i] = v;
}

__device__ __forceinline__ float leaky(float x) {
  return x > 0.f ? x : NEG_SLOPE * x;
}

__device__ __forceinline__ void atomicMaxF(float* addr, float val) {
  unsigned int* ia = (unsigned int*)addr;
  unsigned int old = *ia;
  while (__uint_as_float(old) < val) {
    unsigned int assumed = old;
    old = atomicCAS(ia, assumed, __float_as_uint(val));
    if (old == assumed) break;
  }
}

// ---------------------------------------------------------------------------
// Edge passes. Virtual edges [E, E+N) are the self-loops.
// ---------------------------------------------------------------------------
__global__ void edge_max(const int* __restrict__ srcA, const int* __restrict__ dstA,
                         long long E, int N,
                         const float* __restrict__ a_src, const float* __restrict__ a_dst,
                         float* __restrict__ mbuf, int H) {
  long long ET = E + N;
  long long i = (long long)blockIdx.x * blockDim.x + threadIdx.x;
  if (i >= ET * H) return;
  long long e = i / H; int hh = (int)(i - e * H);
  int s = (e < E) ? srcA[e] : (int)(e - E);
  int d = (e < E) ? dstA[e] : (int)(e - E);
  float ev = leaky(a_src[(size_t)s * H + hh] + a_dst[(size_t)d * H + hh]);
  atomicMaxF(&mbuf[(size_t)d * H + hh], ev);
}

__global__ void edge_sum(const int* __restrict__ srcA, const int* __restrict__ dstA,
                         long long E, int N,
                         const float* __restrict__ a_src, const float* __restrict__ a_dst,
                         const float* __restrict__ mbuf, float* __restrict__ sbuf, int H) {
  long long ET = E + N;
  long long i = (long long)blockIdx.x * blockDim.x + threadIdx.x;
  if (i >= ET * H) return;
  long long e = i / H; int hh = (int)(i - e * H);
  int s = (e < E) ? srcA[e] : (int)(e - E);
  int d = (e < E) ? dstA[e] : (int)(e - E);
  float ev = leaky(a_src[(size_t)s * H + hh] + a_dst[(size_t)d * H + hh]);
  atomicAdd(&sbuf[(size_t)d * H + hh], expf(ev - mbuf[(size_t)d * H + hh]));
}

// One block per edge, blockDim.x == C (channels per head)
__global__ void edge_aggregate(const int* __restrict__ srcA, const int* __restrict__ dstA,
                               long long E, int N,
                               const float* __restrict__ a_src, const float* __restrict__ a_dst,
                               const float* __restrict__ mbuf, const float* __restrict__ sbuf,
                               const float* __restrict__ h, float* __restrict__ out,
                               int H, int C) {
  long long e = blockIdx.x;
  int t = threadIdx.x;
  int s = (e < E) ? srcA[e] : (int)(e - E);
  int d = (e < E) ? dstA[e] : (int)(e - E);
  for (int hh = 0; hh < H; ++hh) {
    float ev = leaky(a_src[(size_t)s * H + hh] + a_dst[(size_t)d * H + hh]);
    float alpha = expf(ev - mbuf[(size_t)d * H + hh]) /
                  (sbuf[(size_t)d * H + hh] + 1e-16f);
    int j = hh * C + t;
    atomicAdd(&out[(size_t)d * H * C + j], alpha * h[(size_t)s * H * C + j]);
  }
}

__global__ void elu_bias(float* __restrict__ out, const float* __restrict__ bias,
                         size_t total, int F) {
  size_t i = (size_t)blockIdx.x * blockDim.x + threadIdx.x;
  if (i >= total) return;
  float v = out[i] + bias[i % (size_t)F];
  out[i] = v > 0.f ? v : (expf(v) - 1.f);
}

// blockDim.x == C (128): bias add, row softmax, optional argmax->float
__global__ void softmax_rows(const float* __restrict__ x, const float* __restrict__ bias,
                             float* __restrict__ out, float* __restrict__ pred, int C) {
  int n = blockIdx.x;
  int t = threadIdx.x;
  __shared__ float red[128];
  __shared__ int   redi[128];
  float v = x[(size_t)n * C + t] + bias[t];
  red[t] = v; redi[t] = t;
  __syncthreads();
  for (int off = 64; off > 0; off >>= 1) {
    if (t < off) {
      float a = red[t], b = red[t + off];
      if (b > a || (b == a && redi[t + off] < redi[t])) {
        red[t] = b; redi[t] = redi[t + off];
      }
    }
    __syncthreads();
  }
  float mx = red[0];
  int amax = redi[0];
  __syncthreads();
  float ex = expf(v - mx);
  red[t] = ex;
  __syncthreads();
  for (int off = 64; off > 0; off >>= 1) {
    if (t < off) red[t] += red[t + off];
    __syncthreads();
  }
  out[(size_t)n * C + t] = ex / red[0];
  if (pred != nullptr && t == 0) pred[n] = (float)amax;
}

// ---------------------------------------------------------------------------
extern "C" void kernel_launch(void* const* d_in, const int* in_sizes, int n_in,
                              void* d_out, int out_size, void* d_ws, size_t ws_size,
                              hipStream_t stream) {
  const float* x        = (const float*)d_in[0];
  const int*   ei1      = (const int*)d_in[1];
  const int*   ei2      = (const int*)d_in[2];
  const float* W1       = (const float*)d_in[3];
  const float* att_s1   = (const float*)d_in[4];
  const float* att_d1   = (const float*)d_in[5];
  const float* b1       = (const float*)d_in[6];
  const float* W2       = (const float*)d_in[7];
  const float* att_s2   = (const float*)d_in[8];
  const float* att_d2   = (const float*)d_in[9];
  const float* b2       = (const float*)d_in[10];
  const float* W3       = (const float*)d_in[11];
  const float* att_s3   = (const float*)d_in[12];
  const float* att_d3   = (const float*)d_in[13];
  const float* b3       = (const float*)d_in[14];

  const int IN = 512, HID = 256, H = 8, F1 = HID * H /*2048*/, OUT = 128;
  const int N = in_sizes[0] / IN;            // 50000
  const long long E = in_sizes[1] / 2;       // 1600000
  const long long ET = E + N;

  // ---- workspace layout ----
  char* ws = (char*)d_ws;
  size_t off = 0;
  auto alloc = [&](size_t bytes) -> float* {
    float* p = (float*)(ws + off);
    off += (bytes + 255) & ~(size_t)255;
    return p;
  };
  float* h1   = alloc((size_t)N * F1 * 4);   // x@W1  (dead after layer-1 aggregate)
  float* emb  = alloc((size_t)N * F1 * 4);   // layer-1 output accumulator -> ELU(emb)
  float* as1  = alloc((size_t)N * H * 4);
  float* ad1  = alloc((size_t)N * H * 4);
  float* m1   = alloc((size_t)N * H * 4);
  float* s1   = alloc((size_t)N * H * 4);
  // layer-2/3 buffers overlay the (dead) h1 region
  float* h2   = h1;
  float* out2 = h1 + 1 * (size_t)N * OUT;
  float* h3   = h1 + 2 * (size_t)N * OUT;
  float* out3 = h1 + 3 * (size_t)N * OUT;
  float* as2  = h1 + 4 * (size_t)N * OUT;
  float* ad2  = as2 + N;
  float* m2   = ad2 + N;
  float* s2   = m2 + N;
  float* as3  = s2 + N;
  float* ad3  = as3 + N;
  float* m3   = ad3 + N;
  float* s3   = m3 + N;

  float* logits1 = (float*)d_out;                          // [N,128]
  float* logits2 = logits1 + (size_t)N * OUT;              // [N,128]
  float* preds   = logits2 + (size_t)N * OUT;              // [N] as float

  const unsigned ebH1 = (unsigned)((ET * H + 255) / 256);
  const unsigned ebH  = (unsigned)((ET + 255) / 256);

  // ================= Layer 1: GAT(512 -> 8x256), concat, ELU =================
  gemm_wmma_f32<<<dim3(N / 16, F1 / 128), 256, 0, stream>>>(x, W1, h1, N, IN, F1);
  attn_coef<<<(N * H + 255) / 256, 256, 0, stream>>>(h1, att_s1, att_d1, as1, ad1, N, H, HID);
  fill_f32<<<(N * H + 255) / 256, 256, 0, stream>>>(m1, -1e30f, (size_t)N * H);
  fill_f32<<<(N * H + 255) / 256, 256, 0, stream>>>(s1, 0.f, (size_t)N * H);
  fill_f32<<<(unsigned)(((size_t)N * F1 + 255) / 256), 256, 0, stream>>>(emb, 0.f, (size_t)N * F1);
  edge_max<<<ebH1, 256, 0, stream>>>(ei1, ei1 + E, E, N, as1, ad1, m1, H);
  edge_sum<<<ebH1, 256, 0, stream>>>(ei1, ei1 + E, E, N, as1, ad1, m1, s1, H);
  edge_aggregate<<<(unsigned)ET, HID, 0, stream>>>(ei1, ei1 + E, E, N, as1, ad1, m1, s1,
                                                   h1, emb, H, HID);
  elu_bias<<<(unsigned)(((size_t)N * F1 + 255) / 256), 256, 0, stream>>>(emb, b1, (size_t)N * F1, F1);

  // ================= Layer 2: GAT(2048 -> 128), heads=1, edge_index ==========
  gemm_wmma_f32<<<dim3(N / 16, OUT / 128), 256, 0, stream>>>(emb, W2, h2, N, F1, OUT);
  attn_coef<<<(N + 255) / 256, 256, 0, stream>>>(h2, att_s2, att_d2, as2, ad2, N, 1, OUT);
  fill_f32<<<(N + 255) / 256, 256, 0, stream>>>(m2, -1e30f, (size_t)N);
  fill_f32<<<(N + 255) / 256, 256, 0, stream>>>(s2, 0.f, (size_t)N);
  fill_f32<<<(unsigned)(((size_t)N * OUT + 255) / 256), 256, 0, stream>>>(out2, 0.f, (size_t)N * OUT);
  edge_max<<<ebH, 256, 0, stream>>>(ei1, ei1 + E, E, N, as2, ad2, m2, 1);
  edge_sum<<<ebH, 256, 0, stream>>>(ei1, ei1 + E, E, N, as2, ad2, m2, s2, 1);
  edge_aggregate<<<(unsigned)ET, OUT, 0, stream>>>(ei1, ei1 + E, E, N, as2, ad2, m2, s2,
                                                   h2, out2, 1, OUT);

  // ================= Layer 3: GAT(2048 -> 128), heads=1, edge_index_2 ========
  gemm_wmma_f32<<<dim3(N / 16, OUT / 128), 256, 0, stream>>>(emb, W3, h3, N, F1, OUT);
  attn_coef<<<(N + 255) / 256, 256, 0, stream>>>(h3, att_s3, att_d3, as3, ad3, N, 1, OUT);
  fill_f32<<<(N + 255) / 256, 256, 0, stream>>>(m3, -1e30f, (size_t)N);
  fill_f32<<<(N + 255) / 256, 256, 0, stream>>>(s3, 0.f, (size_t)N);
  fill_f32<<<(unsigned)(((size_t)N * OUT + 255) / 256), 256, 0, stream>>>(out3, 0.f, (size_t)N * OUT);
  edge_max<<<ebH, 256, 0, stream>>>(ei2, ei2 + E, E, N, as3, ad3, m3, 1);
  edge_sum<<<ebH, 256, 0, stream>>>(ei2, ei2 + E, E, N, as3, ad3, m3, s3, 1);
  edge_aggregate<<<(unsigned)ET, OUT, 0, stream>>>(ei2, ei2 + E, E, N, as3, ad3, m3, s3,
                                                   h3, out3, 1, OUT);

  // ================= Heads: bias + softmax (+ argmax) ========================
  softmax_rows<<<N, OUT, 0, stream>>>(out2, b2, logits1, preds, OUT);
  softmax_rows<<<N, OUT, 0, stream>>>(out3, b3, logits2, nullptr, OUT);
}